// GraphConv_74345883894096
// MI455X (gfx1250) — compile-verified
//
#include <hip/hip_runtime.h>
#include <hip/hip_bf16.h>

// EdgeConv: out[i] = max over edges (j->i) of ReLU([x_i, x_j-x_i] @ W + b), 0 if no edges.
// Restructured: A = x @ (W1-W2) + b ; B = x @ W2 ; msg = ReLU(A[dst] + B[src]).
// Workspace: A (50000*256 f32) + B (50000*256 f32) = 102.4 MB in d_ws.

typedef float v2f __attribute__((ext_vector_type(2)));
typedef float v8f __attribute__((ext_vector_type(8)));

#define IN_CH_  128
#define OUT_CH_ 256

// ---------------------------------------------------------------------------
// Zero-fill output (0x00000000 == 0.0f == identity of relu'd segment-max)
// ---------------------------------------------------------------------------
__global__ void edgeconv_zero_kernel(float* __restrict__ out, int n) {
    int i = blockIdx.x * blockDim.x + threadIdx.x;
    if (i < n) out[i] = 0.0f;
}

// ---------------------------------------------------------------------------
// Dual GEMM via V_WMMA_F32_16X16X4_F32 (full fp32 precision).
//   A[r,c] = sum_k x[r,k] * (W[k,c] - W[k+128,c]) + bias[c]
//   B[r,c] = sum_k x[r,k] *  W[k+128,c]
// Block = 128 threads (4 waves). Wave w computes rows [16*bid,16*bid+16),
// cols [64*w, 64*w+64) of BOTH A and B (x fragment shared by 8 WMMAs/k-step).
// ---------------------------------------------------------------------------
__global__ void __launch_bounds__(128)
edgeconv_gemm_kernel(const float* __restrict__ x,
                     const float* __restrict__ W,
                     const float* __restrict__ bias,
                     float* __restrict__ A,
                     float* __restrict__ B) {
    const int strip   = blockIdx.x;            // 16-node strip
    const int lane    = threadIdx.x & 31;
    const int waveId  = threadIdx.x >> 5;      // 0..3 -> column group
    const int m       = lane & 15;             // row-in-tile / col-in-tile
    const int khalf   = lane >> 4;             // 0: K={0,1}, 1: K={2,3}
    const int row     = strip * 16 + m;
    const int colbase = waveId * 64;

    const float* xrow = x + (size_t)row * IN_CH_;

    v8f accA[4], accB[4];
#pragma unroll
    for (int t = 0; t < 4; ++t) { accA[t] = (v8f)0.0f; accB[t] = (v8f)0.0f; }

    for (int k0 = 0; k0 < IN_CH_; k0 += 4) {
        const int k = k0 + khalf * 2;
        // A-matrix fragment (16x4 fp32): lane holds x[row, k], x[row, k+1]
        v2f a;
        a.x = xrow[k];
        a.y = xrow[k + 1];

#pragma unroll
        for (int t = 0; t < 4; ++t) {
            const int c = colbase + t * 16 + m;
            // B-matrix fragments (4x16 fp32): lane holds W[k, c], W[k+1, c]
            const float w1_0 = W[(size_t)(k)             * OUT_CH_ + c];
            const float w1_1 = W[(size_t)(k + 1)         * OUT_CH_ + c];
            const float w2_0 = W[(size_t)(k + IN_CH_)     * OUT_CH_ + c];
            const float w2_1 = W[(size_t)(k + 1 + IN_CH_) * OUT_CH_ + c];
            v2f wd; wd.x = w1_0 - w2_0; wd.y = w1_1 - w2_1;   // W1 - W2
            v2f w2; w2.x = w2_0;        w2.y = w2_1;          // W2

            accA[t] = __builtin_amdgcn_wmma_f32_16x16x4_f32(
                false, a, false, wd, (short)0, accA[t], false, false);
            accB[t] = __builtin_amdgcn_wmma_f32_16x16x4_f32(
                false, a, false, w2, (short)0, accB[t], false, false);
        }
    }

    // D layout: VGPR v, lane l -> row = v + (l>>4)*8, col = l&15
#pragma unroll
    for (int t = 0; t < 4; ++t) {
        const int c  = colbase + t * 16 + m;
        const float bv = bias[c];
#pragma unroll
        for (int v = 0; v < 8; ++v) {
            const int r = strip * 16 + khalf * 8 + v;
            A[(size_t)r * OUT_CH_ + c] = accA[t][v] + bv;
            B[(size_t)r * OUT_CH_ + c] = accB[t][v];
        }
    }
}

// ---------------------------------------------------------------------------
// Edge phase: one wave per edge. msg = relu(A[dst]+B[src]); segment-max via
// u32 atomicMax (valid: msg >= 0, uint order == float order for >= 0).
// A/B/out together (153 MB) are L2-resident on MI455X (192 MB L2).
// ---------------------------------------------------------------------------
__global__ void __launch_bounds__(256)
edgeconv_edge_kernel(const long long* __restrict__ ei,   // [2, E] int64: row0=src, row1=dst
                     const float* __restrict__ A,
                     const float* __restrict__ B,
                     unsigned int* __restrict__ out,
                     int E) {
    const int gwave = (int)((blockIdx.x * (unsigned)blockDim.x + threadIdx.x) >> 5);
    const int lane  = threadIdx.x & 31;
    if (gwave >= E) return;

    const long long src = ei[gwave];
    const long long dst = ei[(size_t)E + gwave];

    const float4* a4 = (const float4*)(A + (size_t)dst * OUT_CH_);
    const float4* b4 = (const float4*)(B + (size_t)src * OUT_CH_);
    unsigned int* o  = out + (size_t)dst * OUT_CH_;

#pragma unroll
    for (int r = 0; r < 2; ++r) {
        const int idx = lane + r * 32;          // float4 index 0..63
        const float4 av = a4[idx];
        const float4 bv = b4[idx];
        const float m0 = fmaxf(av.x + bv.x, 0.0f);
        const float m1 = fmaxf(av.y + bv.y, 0.0f);
        const float m2 = fmaxf(av.z + bv.z, 0.0f);
        const float m3 = fmaxf(av.w + bv.w, 0.0f);
        atomicMax(o + idx * 4 + 0, __float_as_uint(m0));
        atomicMax(o + idx * 4 + 1, __float_as_uint(m1));
        atomicMax(o + idx * 4 + 2, __float_as_uint(m2));
        atomicMax(o + idx * 4 + 3, __float_as_uint(m3));
    }
}

// ---------------------------------------------------------------------------
extern "C" void kernel_launch(void* const* d_in, const int* in_sizes, int n_in,
                              void* d_out, int out_size, void* d_ws, size_t ws_size,
                              hipStream_t stream) {
    const float*     x  = (const float*)d_in[0];      // [N, 128] f32
    const long long* ei = (const long long*)d_in[1];  // [2, E]  i64
    const float*     W  = (const float*)d_in[2];      // [256, 256] f32
    const float*     b  = (const float*)d_in[3];      // [256] f32
    float*           out = (float*)d_out;             // [N, 256] f32

    const int nNodes = in_sizes[0] / IN_CH_;          // 50000
    const int E      = in_sizes[1] / 2;               // 800000

    float* A = (float*)d_ws;
    float* B = A + (size_t)nNodes * OUT_CH_;

    // 1) out = 0 (identity for relu'd segment-max; also covers edge-less nodes)
    const int outN = nNodes * OUT_CH_;
    edgeconv_zero_kernel<<<(outN + 255) / 256, 256, 0, stream>>>(out, outN);

    // 2) A = x @ (W1 - W2) + b ; B = x @ W2   (50000 = 16 * 3125 exactly)
    const int strips = (nNodes + 15) / 16;
    edgeconv_gemm_kernel<<<strips, 128, 0, stream>>>(x, W, b, A, B);

    // 3) per-edge relu-add + atomic segment-max (8 waves / 256-thread block)
    const int blocks = (E + 7) / 8;
    edgeconv_edge_kernel<<<blocks, 256, 0, stream>>>(ei, A, B,
                                                     (unsigned int*)out, E);
}